// CAN_Layer_85822036509221
// MI455X (gfx1250) — compile-verified
//
#include <hip/hip_runtime.h>
#include <hip/hip_bf16.h>

typedef __attribute__((ext_vector_type(2))) float v2f;
typedef __attribute__((ext_vector_type(8))) float v8f;

constexpr int cN  = 16;
constexpr int cD  = 256;
constexpr int cH  = 8;
constexpr int cHD = 32;
constexpr int cGS = 4;
constexpr int cGP = 512;   // LP/GS
constexpr int cGD = 128;   // LD/GS

// ---------------------------------------------------------------------------
// 1) Group means + group masks.  x:[N, G*GS, D] -> xg:[N,G,D], mg:[N,G] (0/1)
// ---------------------------------------------------------------------------
__global__ void group_kernel(const float* __restrict__ x,
                             const unsigned char* __restrict__ m,
                             float* __restrict__ xg, float* __restrict__ mg, int G) {
  long idx = (long)blockIdx.x * blockDim.x + threadIdx.x;  // over N*G*D
  long total = (long)cN * G * cD;
  if (idx >= total) return;
  int  k  = (int)(idx % cD);
  long ng = idx / cD;                       // n*G + g
  long base = ng * (long)cGS * cD + k;
  float s = x[base] + x[base + cD] + x[base + 2 * cD] + x[base + 3 * cD];
  xg[idx] = 0.25f * s;
  if (k == 0) {
    long mb = ng * cGS;
    unsigned char v = (unsigned char)(m[mb] | m[mb + 1] | m[mb + 2] | m[mb + 3]);
    mg[ng] = v ? 1.0f : 0.0f;
  }
}

// ---------------------------------------------------------------------------
// 2) Projection GEMM: Y[M,256] = X[M,256] @ W^T  via V_WMMA_F32_16X16X4_F32
//    one wave per 16x16 output tile, K swept in steps of 4.
// ---------------------------------------------------------------------------
__global__ void proj_wmma(const float* __restrict__ X, const float* __restrict__ W,
                          float* __restrict__ Y, int M) {
  long wave = ((long)blockIdx.x * blockDim.x + threadIdx.x) >> 5;
  int  lane = threadIdx.x & 31;
  const int tilesN = cD / 16;               // 16
  long totalWaves = (long)(M / 16) * tilesN;
  if (wave >= totalWaves) return;           // uniform per wave
  int tileM = (int)(wave / tilesN);
  int tileN = (int)(wave % tilesN);
  int row0 = tileM * 16, col0 = tileN * 16;
  int lm = lane & 15, lh = lane >> 4;       // lh in {0,1}
  const float* xrow = X + (long)(row0 + lm) * cD;
  const float* wrow = W + (long)(col0 + lm) * cD;
  v8f acc = {};
  for (int k = 0; k < cD; k += 4) {
    v2f a, b;
    a.x = xrow[k + 2 * lh]; a.y = xrow[k + 2 * lh + 1];
    b.x = wrow[k + 2 * lh]; b.y = wrow[k + 2 * lh + 1];
    acc = __builtin_amdgcn_wmma_f32_16x16x4_f32(false, a, false, b,
                                                (short)0, acc, false, false);
  }
  for (int v = 0; v < 8; ++v)
    Y[(long)(row0 + v + 8 * lh) * cD + col0 + lm] = acc[v];
}

// ---------------------------------------------------------------------------
// 3) Attention logits per (n,h): out[l1,l2] = (Q.K^T)/sqrt(32), -1e6 if !pair
//    out layout: [n][h][L1][L2]
// ---------------------------------------------------------------------------
__global__ void logits_wmma(const float* __restrict__ Q, const float* __restrict__ K,
                            const float* __restrict__ mrow, const float* __restrict__ mcol,
                            float* __restrict__ out, int L1, int L2) {
  long wave = ((long)blockIdx.x * blockDim.x + threadIdx.x) >> 5;
  int  lane = threadIdx.x & 31;
  int t1 = L1 >> 4, t2 = L2 >> 4;
  long tilesPerNH = (long)t1 * t2;
  long nh = wave / tilesPerNH;
  if (nh >= (long)cN * cH) return;          // uniform per wave
  long rem = wave % tilesPerNH;
  int tm = (int)(rem / t2), tn = (int)(rem % t2);
  int n = (int)(nh / cH), h = (int)(nh % cH);
  int lm = lane & 15, lh = lane >> 4;
  const float* qrow = Q + ((long)n * L1 + tm * 16 + lm) * cD + h * cHD;
  const float* krow = K + ((long)n * L2 + tn * 16 + lm) * cD + h * cHD;
  v8f acc = {};
  for (int k = 0; k < cHD; k += 4) {
    v2f a, b;
    a.x = qrow[k + 2 * lh]; a.y = qrow[k + 2 * lh + 1];
    b.x = krow[k + 2 * lh]; b.y = krow[k + 2 * lh + 1];
    acc = __builtin_amdgcn_wmma_f32_16x16x4_f32(false, a, false, b,
                                                (short)0, acc, false, false);
  }
  const float scale = 0.17677669529663687f; // 1/sqrt(32)
  int l2i = tn * 16 + lm;
  float mc = mcol[(long)n * L2 + l2i];
  float* obase = out + (long)nh * L1 * L2;
  for (int v = 0; v < 8; ++v) {
    int l1i = tm * 16 + v + 8 * lh;
    float val = acc[v] * scale;
    float mr = mrow[(long)n * L1 + l1i];
    if (!(mr > 0.5f && mc > 0.5f)) val -= 1.0e6f;
    obase[(long)l1i * L2 + l2i] = val;
  }
}

// ---------------------------------------------------------------------------
// 4) Global softmax over L1*L2 per (n,h), in place:
//    A0 = valid ? softmax + 1e-12 : 0
// ---------------------------------------------------------------------------
__global__ void softmax_kernel(float* __restrict__ A, const float* __restrict__ mrow,
                               const float* __restrict__ mcol, int L1, int L2) {
  int nh = blockIdx.x;
  int n  = nh / cH;
  long M = (long)L1 * L2;
  float* base = A + (long)nh * M;
  __shared__ float red[256];
  int t = threadIdx.x;
  float mx = -3.4e38f;
  for (long i = t; i < M; i += 256) mx = fmaxf(mx, base[i]);
  red[t] = mx; __syncthreads();
  for (int s = 128; s > 0; s >>= 1) { if (t < s) red[t] = fmaxf(red[t], red[t + s]); __syncthreads(); }
  float gmax = red[0]; __syncthreads();
  float sum = 0.f;
  for (long i = t; i < M; i += 256) sum += __expf(base[i] - gmax);
  red[t] = sum; __syncthreads();
  for (int s = 128; s > 0; s >>= 1) { if (t < s) red[t] += red[t + s]; __syncthreads(); }
  float inv = 1.0f / red[0];
  const float* mr = mrow + (long)n * L1;
  const float* mc = mcol + (long)n * L2;
  for (long i = t; i < M; i += 256) {
    long l1 = i / L2; long l2 = i - l1 * L2;
    bool valid = (mr[l1] > 0.5f) && (mc[l2] > 0.5f);
    base[i] = valid ? (__expf(base[i] - gmax) * inv + 1e-12f) : 0.0f;
  }
}

// ---------------------------------------------------------------------------
// 5) Sinkhorn as diagonal scaling: A_t = diag(r) A0 diag(c).
//    4 iterations of (r update via row matvec, c update via col matvec),
//    then final column sums w[j] = c[j] * sum_i A0[i,j] r[i].
// ---------------------------------------------------------------------------
__global__ void sinkhorn_kernel(const float* __restrict__ A, float* __restrict__ colw,
                                float* __restrict__ rout, float* __restrict__ cout,
                                int L1, int L2) {
  int nh = blockIdx.x;
  const float* base = A + (long)nh * L1 * L2;
  __shared__ float r[512];
  __shared__ float c[512];
  int t = threadIdx.x;
  for (int i = t; i < L1; i += 256) r[i] = 1.0f;
  for (int j = t; j < L2; j += 256) c[j] = 1.0f;
  __syncthreads();
  int wv = t >> 5, lane = t & 31;           // 8 waves of 32
  for (int it = 0; it < 4; ++it) {
    for (int i = wv; i < L1; i += 8) {      // row pass: one wave per row
      const float* row = base + (long)i * L2;
      float s = 0.f;
      for (int j = lane; j < L2; j += 32) s += row[j] * c[j];
      for (int o = 16; o > 0; o >>= 1) s += __shfl_down(s, o);
      if (lane == 0) { float ri = r[i]; r[i] = ri / (ri * s + 1e-12f); }
    }
    __syncthreads();
    for (int j = t; j < L2; j += 256) {     // col pass
      float s = 0.f;
      for (int i = 0; i < L1; ++i) s += base[(long)i * L2 + j] * r[i];
      float cj = c[j];
      c[j] = cj / (cj * s + 1e-12f);
    }
    __syncthreads();
  }
  for (int j = t; j < L2; j += 256) {       // final column sums of A_final
    float s = 0.f;
    for (int i = 0; i < L1; ++i) s += base[(long)i * L2 + j] * r[i];
    colw[(long)nh * L2 + j] = c[j] * s;
  }
  if (rout) for (int i = t; i < L1; i += 256) rout[(long)nh * L1 + i] = r[i];
  if (cout) for (int j = t; j < L2; j += 256) cout[(long)nh * L2 + j] = c[j];
}

// ---------------------------------------------------------------------------
// 6) Materialize a_dp output: [n][h][l1][l2] * r * c  ->  [n][l1][l2][h]
// ---------------------------------------------------------------------------
__global__ void adp_out_kernel(const float* __restrict__ A, const float* __restrict__ r,
                               const float* __restrict__ c, float* __restrict__ out) {
  long idx = (long)blockIdx.x * blockDim.x + threadIdx.x;
  long total = (long)cN * cH * cGD * cGP;
  if (idx >= total) return;
  int  l2 = (int)(idx % cGP);
  long q  = idx / cGP;
  int  l1 = (int)(q % cGD); q /= cGD;
  int  h  = (int)(q % cH);
  int  n  = (int)(q / cH);
  float v = A[idx] * r[((long)n * cH + h) * cGD + l1]
                   * c[((long)n * cH + h) * cGP + l2];
  out[(((long)n * cGD + l1) * cGP + l2) * cH + h] = v;
}

// ---------------------------------------------------------------------------
// 7) Masked-mean embeddings collapse to colsum-weighted V:
//    prot_embed[n,j] = 0.5/cntP * (sum_k w_pp[h,k] vp[k,j] + sum_k w_pd[h,k] vd[k,j])
// ---------------------------------------------------------------------------
__global__ void embed_kernel(const float* __restrict__ vp, const float* __restrict__ vd,
                             const float* __restrict__ wpp, const float* __restrict__ wpd,
                             const float* __restrict__ wdp, const float* __restrict__ wdd,
                             const float* __restrict__ mp, const float* __restrict__ md,
                             float* __restrict__ out) {
  int n = blockIdx.x;
  int j = threadIdx.x;          // 0..255
  int h = j >> 5;               // j / HD
  __shared__ float redP[256], redD[256];
  float cp = 0.f, cd = 0.f;
  for (int g = j; g < cGP; g += 256) cp += mp[n * cGP + g];
  for (int g = j; g < cGD; g += 256) cd += md[n * cGD + g];
  redP[j] = cp; redD[j] = cd; __syncthreads();
  for (int s = 128; s > 0; s >>= 1) {
    if (j < s) { redP[j] += redP[j + s]; redD[j] += redD[j + s]; }
    __syncthreads();
  }
  cp = redP[0]; cd = redD[0];
  float accP = 0.f, accD = 0.f;
  const float* wa = wpp + ((long)n * cH + h) * cGP;
  const float* wb = wpd + ((long)n * cH + h) * cGD;
  const float* wc = wdp + ((long)n * cH + h) * cGP;
  const float* wd = wdd + ((long)n * cH + h) * cGD;
  for (int k = 0; k < cGP; ++k) {
    float v = vp[((long)n * cGP + k) * cD + j];
    accP += wa[k] * v;
    accD += wc[k] * v;
  }
  for (int k = 0; k < cGD; ++k) {
    float v = vd[((long)n * cGD + k) * cD + j];
    accP += wb[k] * v;
    accD += wd[k] * v;
  }
  out[(long)n * 2 * cD + j]      = 0.5f * accP / cp;
  out[(long)n * 2 * cD + cD + j] = 0.5f * accD / cd;
}

// ---------------------------------------------------------------------------
extern "C" void kernel_launch(void* const* d_in, const int* in_sizes, int n_in,
                              void* d_out, int out_size, void* d_ws, size_t ws_size,
                              hipStream_t stream) {
  (void)in_sizes; (void)n_in; (void)out_size; (void)ws_size;
  const float*         protein = (const float*)d_in[0];
  const float*         drug    = (const float*)d_in[1];
  const unsigned char* mask_p  = (const unsigned char*)d_in[2];
  const unsigned char* mask_d  = (const unsigned char*)d_in[3];
  const float* Wq_p = (const float*)d_in[4];
  const float* Wk_p = (const float*)d_in[5];
  const float* Wv_p = (const float*)d_in[6];
  const float* Wq_d = (const float*)d_in[7];
  const float* Wk_d = (const float*)d_in[8];
  const float* Wv_d = (const float*)d_in[9];
  float* out = (float*)d_out;

  float* ws = (float*)d_ws;
  float* pg  = ws; ws += (long)cN * cGP * cD;
  float* dg  = ws; ws += (long)cN * cGD * cD;
  float* mpF = ws; ws += (long)cN * cGP;
  float* mdF = ws; ws += (long)cN * cGD;
  float* qp  = ws; ws += (long)cN * cGP * cD;
  float* kp  = ws; ws += (long)cN * cGP * cD;
  float* vp  = ws; ws += (long)cN * cGP * cD;
  float* qd  = ws; ws += (long)cN * cGD * cD;
  float* kd  = ws; ws += (long)cN * cGD * cD;
  float* vd  = ws; ws += (long)cN * cGD * cD;
  float* App = ws; ws += (long)cN * cH * cGP * cGP;
  float* Apd = ws; ws += (long)cN * cH * cGP * cGD;
  float* Adp = ws; ws += (long)cN * cH * cGD * cGP;
  float* Add = ws; ws += (long)cN * cH * cGD * cGD;
  float* wpp = ws; ws += (long)cN * cH * cGP;
  float* wpd = ws; ws += (long)cN * cH * cGD;
  float* wdp = ws; ws += (long)cN * cH * cGP;
  float* wdd = ws; ws += (long)cN * cH * cGD;
  float* rdp = ws; ws += (long)cN * cH * cGD;
  float* cdp = ws; ws += (long)cN * cH * cGP;

  // 1) grouping
  group_kernel<<<(cN * cGP * cD) / 256, 256, 0, stream>>>(protein, mask_p, pg, mpF, cGP);
  group_kernel<<<(cN * cGD * cD) / 256, 256, 0, stream>>>(drug,    mask_d, dg, mdF, cGD);

  // 2) projections (waves == M; 4 waves per 128-thread block)
  const int Mp = cN * cGP, Md = cN * cGD;
  proj_wmma<<<Mp / 4, 128, 0, stream>>>(pg, Wq_p, qp, Mp);
  proj_wmma<<<Mp / 4, 128, 0, stream>>>(pg, Wk_p, kp, Mp);
  proj_wmma<<<Mp / 4, 128, 0, stream>>>(pg, Wv_p, vp, Mp);
  proj_wmma<<<Md / 4, 128, 0, stream>>>(dg, Wq_d, qd, Md);
  proj_wmma<<<Md / 4, 128, 0, stream>>>(dg, Wk_d, kd, Md);
  proj_wmma<<<Md / 4, 128, 0, stream>>>(dg, Wv_d, vd, Md);

  // 3) logits (blocks = waves/4)
  auto lgBlocks = [](int L1, int L2) { return (cN * cH * (L1 / 16) * (L2 / 16)) / 4; };
  logits_wmma<<<lgBlocks(cGP, cGP), 128, 0, stream>>>(qp, kp, mpF, mpF, App, cGP, cGP);
  logits_wmma<<<lgBlocks(cGP, cGD), 128, 0, stream>>>(qp, kd, mpF, mdF, Apd, cGP, cGD);
  logits_wmma<<<lgBlocks(cGD, cGP), 128, 0, stream>>>(qd, kp, mdF, mpF, Adp, cGD, cGP);
  logits_wmma<<<lgBlocks(cGD, cGD), 128, 0, stream>>>(qd, kd, mdF, mdF, Add, cGD, cGD);

  // 4) softmax (in place -> A0)
  softmax_kernel<<<cN * cH, 256, 0, stream>>>(App, mpF, mpF, cGP, cGP);
  softmax_kernel<<<cN * cH, 256, 0, stream>>>(Apd, mpF, mdF, cGP, cGD);
  softmax_kernel<<<cN * cH, 256, 0, stream>>>(Adp, mdF, mpF, cGD, cGP);
  softmax_kernel<<<cN * cH, 256, 0, stream>>>(Add, mdF, mdF, cGD, cGD);

  // 5) sinkhorn (diagonal-scaling form) + final column sums
  sinkhorn_kernel<<<cN * cH, 256, 0, stream>>>(App, wpp, nullptr, nullptr, cGP, cGP);
  sinkhorn_kernel<<<cN * cH, 256, 0, stream>>>(Apd, wpd, nullptr, nullptr, cGP, cGD);
  sinkhorn_kernel<<<cN * cH, 256, 0, stream>>>(Adp, wdp, rdp, cdp, cGD, cGP);
  sinkhorn_kernel<<<cN * cH, 256, 0, stream>>>(Add, wdd, nullptr, nullptr, cGD, cGD);

  // 6) a_dp output (after the query_embed block of N*2D floats)
  adp_out_kernel<<<(cN * cH * cGD * cGP) / 256, 256, 0, stream>>>(
      Adp, rdp, cdp, out + (long)cN * 2 * cD);

  // 7) pooled embeddings
  embed_kernel<<<cN, 256, 0, stream>>>(vp, vd, wpp, wpd, wdp, wdd, mpF, mdF, out);
}